// QuantumLayer_71760313582074
// MI455X (gfx1250) — compile-verified
//
#include <hip/hip_runtime.h>

// CDNA5 (gfx1250) wave32 WMMA outer-product kernel, v2.
// out[i,j] = cos(pi*x[i,0]) * cos(theta[0]) * W[j] + b[j]
// D = A(16x4) x B(4x16) + C via V_WMMA_F32_16X16X4_F32 (f32 exact),
// K=0 carries the rank-1 data, K=1..3 zero-padded.
// Each wave reuses one A matrix across 4 N-tiles (4 WMMAs, 64x16 f32 out).

typedef __attribute__((ext_vector_type(2))) float v2f;
typedef __attribute__((ext_vector_type(8))) float v8f;

#define PI_F 3.14159265358979323846f
#define NT 4   // N-tiles per wave (shared A matrix)

__global__ __launch_bounds__(256) void quantum_outer_wmma(
    const float* __restrict__ x,      // (Brows, H), only column 0 used
    const float* __restrict__ theta,  // (H,), only theta[0] used
    const float* __restrict__ W,      // (1, H)
    const float* __restrict__ b,      // (H,)
    float* __restrict__ out,          // (Brows, H)
    int H, int ngroups_n, int total_waves)
{
    const int lane    = threadIdx.x & 31;
    const int l15     = lane & 15;
    const int half    = (lane >> 4) & 1;          // 0: lanes 0-15, 1: lanes 16-31
    const int wave_id = blockIdx.x * (blockDim.x >> 5) + (threadIdx.x >> 5);
    if (wave_id >= total_waves) return;           // wave-uniform; EXEC stays full

    const int grp    = wave_id % ngroups_n;       // group of NT consecutive N-tiles
    const int tile_m = wave_id / ngroups_n;       // consecutive waves share tile_m

    // z[m] = cos(pi*x[m,0]) * cos(theta[0]); __cosf -> v_cos_f32 (hw range
    // reduction), fine for pi*N(0,1) args and avoids libm's Payne-Hanek bloat.
    const float ct  = __cosf(theta[0]);
    const int   row = tile_m * 16 + l15;          // lanes 16-31 read same addrs
    const float xv  = x[(size_t)row * (size_t)H];
    const float zv  = __cosf(PI_F * xv) * ct;

    // A-matrix 16x4 f32: VGPR0 = {K=0 (lanes 0-15), K=2 (lanes 16-31)}, VGPR1 = {K=1,K=3}
    v2f a;
    a.x = half ? 0.0f : zv;                       // branch-free select, EXEC untouched
    a.y = 0.0f;

    const int col0 = grp * (16 * NT) + l15;

    v8f d[NT];
#pragma unroll
    for (int t = 0; t < NT; ++t) {
        const int col  = col0 + 16 * t;
        const float wv   = W[col];
        const float bias = b[col];

        // B-matrix 4x16 f32: row K=0 = W (lanes 0-15), all other K rows zero.
        v2f bm;
        bm.x = half ? 0.0f : wv;
        bm.y = 0.0f;

        // C: bias per column broadcast down all 16 rows (VGPR r: M=r / M=r+8, N=lane%16).
        v8f c;
#pragma unroll
        for (int r = 0; r < 8; ++r) c[r] = bias;

        d[t] = __builtin_amdgcn_wmma_f32_16x16x4_f32(
            /*neg_a=*/false, a, /*neg_b=*/false, bm,
            /*c_mod=*/(short)0, c, /*reuse_a=*/false, /*reuse_b=*/false);
    }

    // Stores grouped by row: 4 consecutive b32 stores cover 256B contiguous
    // per row (per half-wave) -> good write-combining into 128B lines.
    const size_t rb = (size_t)(tile_m * 16 + half * 8);
#pragma unroll
    for (int r = 0; r < 8; ++r) {
        float* rowp = out + (rb + (size_t)r) * (size_t)H + (size_t)col0;
#pragma unroll
        for (int t = 0; t < NT; ++t)
            rowp[16 * t] = d[t][r];
    }
}

extern "C" void kernel_launch(void* const* d_in, const int* in_sizes, int n_in,
                              void* d_out, int out_size, void* d_ws, size_t ws_size,
                              hipStream_t stream) {
    const float* x     = (const float*)d_in[0];
    const float* theta = (const float*)d_in[1];
    const float* W     = (const float*)d_in[2];
    const float* b     = (const float*)d_in[3];
    float* out = (float*)d_out;

    const int H         = in_sizes[3];           // 1024 (bias length)
    const int Brows     = in_sizes[0] / H;       // 32768
    const int mt        = Brows / 16;            // 2048 M-tiles
    const int ngroups_n = H / (16 * NT);         // 16 groups of 4 N-tiles
    const int total_waves = mt * ngroups_n;      // 32768 waves, 4 KB out each

    const int threads = 256;                     // 8 waves per block
    const int wavesPerBlock = threads / 32;
    const int blocks = (total_waves + wavesPerBlock - 1) / wavesPerBlock;

    quantum_outer_wmma<<<blocks, threads, 0, stream>>>(
        x, theta, W, b, out, H, ngroups_n, total_waves);
}